// ComformerConvEqui_50276887166992
// MI455X (gfx1250) — compile-verified
//
#include <hip/hip_runtime.h>

// ============================================================================
// Problem constants
// ============================================================================
#define N_NODES 30000
#define N_EDGES 360000

typedef __attribute__((ext_vector_type(2))) float v2f;
typedef __attribute__((ext_vector_type(8))) float v8f;
typedef __attribute__((ext_vector_type(4))) int   v4i;

// Async global->LDS path (CDNA5, tracked by ASYNCcnt), guarded so the file
// compiles even if the toolchain spells these differently.
#if defined(__gfx1250__) && __has_builtin(__builtin_amdgcn_global_load_async_to_lds_b128) && \
    __has_builtin(__builtin_amdgcn_s_wait_asynccnt)
#define HAVE_ASYNC_LDS 1
#endif

// ============================================================================
// Compile-time e3nn machinery: factorials, sqrt, complex, SU(2) CG, real
// change-of-basis q, real Wigner-3j (normalized), and flattened tensor-product
// entry tables (counting-sorted by output offset for run-length accumulation).
// ============================================================================
constexpr int imax2(int a, int b) { return a > b ? a : b; }
constexpr int imin2(int a, int b) { return a < b ? a : b; }

constexpr double cfact(int n) {
    double r = 1.0;
    for (int i = 2; i <= n; ++i) r *= (double)i;
    return r;
}
constexpr double csqrt(double x) {
    if (x <= 0.0) return 0.0;
    double g = x < 1.0 ? 1.0 : x;
    for (int i = 0; i < 64; ++i) g = 0.5 * (g + x / g);
    return g;
}

struct CPX { double re; double im; };
constexpr CPX cmul(CPX a, CPX b) { return CPX{a.re*b.re - a.im*b.im, a.re*b.im + a.im*b.re}; }
constexpr CPX cconj(CPX a) { return CPX{a.re, -a.im}; }

constexpr double su2_cg(int j1, int m1, int j2, int m2, int j3, int m3) {
    if (m3 != m1 + m2) return 0.0;
    int vmin = imax2(imax2(-j1 + j2 + m3, -j1 + m1), 0);
    int vmax = imin2(imin2(j2 + j3 + m1, j3 - j1 + j2), j3 + m3);
    if (vmax < vmin) return 0.0;
    double c = csqrt((2.0*j3 + 1.0) * cfact(j3 + j1 - j2) * cfact(j3 - j1 + j2) *
                     cfact(j1 + j2 - j3) * cfact(j3 + m3) * cfact(j3 - m3) /
                     (cfact(j1 + j2 + j3 + 1) * cfact(j1 - m1) * cfact(j1 + m1) *
                      cfact(j2 - m2) * cfact(j2 + m2)));
    double s = 0.0;
    for (int v = vmin; v <= vmax; ++v) {
        double sgn = ((v + j2 + m2) % 2 == 0) ? 1.0 : -1.0;
        s += sgn * cfact(j2 + j3 + m1 - v) * cfact(j1 - m1 + v) /
             (cfact(v) * cfact(j3 - j1 + j2 - v) * cfact(j3 + m3 - v) *
              cfact(v + j1 - j2 - m3));
    }
    return c * s;
}

struct QM { CPX q[7][7]; };

constexpr QM make_q(int l) {
    QM Q{};
    const double s = csqrt(0.5);
    for (int m = -l; m < 0; ++m) {
        Q.q[l + m][l - m] = CPX{s, 0.0};
        Q.q[l + m][l + m] = CPX{0.0, -s};
    }
    Q.q[l][l] = CPX{1.0, 0.0};
    for (int m = 1; m <= l; ++m) {
        double sgn = (m % 2 == 0) ? 1.0 : -1.0;
        Q.q[l + m][l + m] = CPX{sgn * s, 0.0};
        Q.q[l + m][l - m] = CPX{0.0, sgn * s};
    }
    CPX ph = CPX{1.0, 0.0};
    int lm = l % 4;
    if (lm == 1) ph = CPX{0.0, -1.0};
    else if (lm == 2) ph = CPX{-1.0, 0.0};
    else if (lm == 3) ph = CPX{0.0, 1.0};
    for (int i = 0; i < 2*l + 1; ++i)
        for (int j = 0; j < 2*l + 1; ++j)
            Q.q[i][j] = cmul(ph, Q.q[i][j]);
    return Q;
}

struct W3J { double v[7][7][7]; };

constexpr W3J make_w3j(int l1, int l2, int l3) {
    const int d1 = 2*l1 + 1, d2 = 2*l2 + 1, d3 = 2*l3 + 1;
    double cg[7][7][7]{};
    for (int m1 = -l1; m1 <= l1; ++m1)
        for (int m2 = -l2; m2 <= l2; ++m2)
            for (int m3 = -l3; m3 <= l3; ++m3)
                cg[l1 + m1][l2 + m2][l3 + m3] = su2_cg(l1, m1, l2, m2, l3, m3);
    QM q1 = make_q(l1), q2 = make_q(l2), q3 = make_q(l3);
    double cr[7][7][7]{}, ci[7][7][7]{};
    for (int i = 0; i < d1; ++i)
        for (int k = 0; k < d2; ++k)
            for (int n = 0; n < d3; ++n) {
                double g = cg[i][k][n];
                if (g == 0.0) continue;
                for (int j = 0; j < d1; ++j) {
                    CPX a = q1.q[i][j];
                    if (a.re == 0.0 && a.im == 0.0) continue;
                    for (int ll = 0; ll < d2; ++ll) {
                        CPX b = q2.q[k][ll];
                        if (b.re == 0.0 && b.im == 0.0) continue;
                        CPX ab = cmul(a, b);
                        for (int m = 0; m < d3; ++m) {
                            CPX cc = cconj(q3.q[n][m]);
                            if (cc.re == 0.0 && cc.im == 0.0) continue;
                            CPX t = cmul(ab, cc);
                            cr[j][ll][m] += g * t.re;
                            ci[j][ll][m] += g * t.im;
                        }
                    }
                }
            }
    double nr = 0.0, ni = 0.0;
    for (int a = 0; a < d1; ++a)
        for (int b = 0; b < d2; ++b)
            for (int c = 0; c < d3; ++c) {
                nr += cr[a][b][c] * cr[a][b][c];
                ni += ci[a][b][c] * ci[a][b][c];
            }
    bool useR = (csqrt(nr) >= csqrt(ni));
    double nrm = csqrt(useR ? nr : ni);
    W3J w{};
    for (int a = 0; a < d1; ++a)
        for (int b = 0; b < d2; ++b)
            for (int c = 0; c < d3; ++c)
                w.v[a][b][c] = (useR ? cr[a][b][c] : ci[a][b][c]) / nrm;
    return w;
}

constexpr W3J W000 = make_w3j(0,0,0);
constexpr W3J W011 = make_w3j(0,1,1);
constexpr W3J W022 = make_w3j(0,2,2);
constexpr W3J W101 = make_w3j(1,0,1);
constexpr W3J W110 = make_w3j(1,1,0);
constexpr W3J W111 = make_w3j(1,1,1);
constexpr W3J W112 = make_w3j(1,1,2);
constexpr W3J W121 = make_w3j(1,2,1);
constexpr W3J W122 = make_w3j(1,2,2);
constexpr W3J W123 = make_w3j(1,2,3);
constexpr W3J W202 = make_w3j(2,0,2);
constexpr W3J W211 = make_w3j(2,1,1);
constexpr W3J W212 = make_w3j(2,1,2);
constexpr W3J W213 = make_w3j(2,1,3);
constexpr W3J W220 = make_w3j(2,2,0);
constexpr W3J W221 = make_w3j(2,2,1);
constexpr W3J W222 = make_w3j(2,2,2);
constexpr W3J W223 = make_w3j(2,2,3);

constexpr const W3J& pick_w3j(int a, int b, int c) {
    if (a==0&&b==0&&c==0) return W000;
    if (a==0&&b==1&&c==1) return W011;
    if (a==0&&b==2&&c==2) return W022;
    if (a==1&&b==0&&c==1) return W101;
    if (a==1&&b==1&&c==0) return W110;
    if (a==1&&b==1&&c==1) return W111;
    if (a==1&&b==1&&c==2) return W112;
    if (a==1&&b==2&&c==1) return W121;
    if (a==1&&b==2&&c==2) return W122;
    if (a==1&&b==2&&c==3) return W123;
    if (a==2&&b==0&&c==2) return W202;
    if (a==2&&b==1&&c==1) return W211;
    if (a==2&&b==1&&c==2) return W212;
    if (a==2&&b==1&&c==3) return W213;
    if (a==2&&b==2&&c==0) return W220;
    if (a==2&&b==2&&c==1) return W221;
    if (a==2&&b==2&&c==2) return W222;
    if (a==2&&b==2&&c==3) return W223;
    return W000;
}

struct Irrep { int mul, l, p; };

struct TPEntry { short o, w, a, s; float c; };
constexpr int MAX_ENT = 4096;
struct Table { int count; int wn; int overflow; TPEntry e[MAX_ENT]; };

template <int N1, int NO>
constexpr Table build_table(const Irrep (&in1)[N1], const Irrep (&out)[NO]) {
    constexpr Irrep in2[3] = {{1,0,1},{1,1,-1},{1,2,1}};
    int s1[8]{}, so[8]{};
    const int s2[3] = {0, 1, 4};
    { int o = 0; for (int i = 0; i < N1; ++i) { s1[i] = o; o += in1[i].mul * (2*in1[i].l + 1); } }
    { int o = 0; for (int i = 0; i < NO; ++i) { so[i] = o; o += out[i].mul * (2*out[i].l + 1); } }
    int fan[8]{};
    for (int i1 = 0; i1 < N1; ++i1)
        for (int i2 = 0; i2 < 3; ++i2)
            for (int io = 0; io < NO; ++io) {
                int l1 = in1[i1].l, l2 = in2[i2].l, l3 = out[io].l;
                if (in1[i1].p * in2[i2].p != out[io].p) continue;
                int lo = l1 > l2 ? l1 - l2 : l2 - l1;
                if (l3 < lo || l3 > l1 + l2) continue;
                fan[io] += in1[i1].mul * in2[i2].mul;
            }
    Table T{};
    int wo = 0;
    for (int i1 = 0; i1 < N1; ++i1)
        for (int i2 = 0; i2 < 3; ++i2)
            for (int io = 0; io < NO; ++io) {
                int l1 = in1[i1].l, l2 = in2[i2].l, l3 = out[io].l;
                if (in1[i1].p * in2[i2].p != out[io].p) continue;
                int lo = l1 > l2 ? l1 - l2 : l2 - l1;
                if (l3 < lo || l3 > l1 + l2) continue;
                int m1 = in1[i1].mul, m3 = out[io].mul;
                const W3J& w3 = pick_w3j(l1, l2, l3);
                double alpha = csqrt((double)(2*l3 + 1) / (double)fan[io]);
                for (int u = 0; u < m1; ++u)
                    for (int w = 0; w < m3; ++w)
                        for (int i = 0; i < 2*l1 + 1; ++i)
                            for (int j = 0; j < 2*l2 + 1; ++j)
                                for (int k = 0; k < 2*l3 + 1; ++k) {
                                    double cf = alpha * w3.v[i][j][k];
                                    if (cf > -1e-12 && cf < 1e-12) continue;
                                    if (T.count >= MAX_ENT) { T.overflow = 1; continue; }
                                    T.e[T.count] = TPEntry{
                                        (short)(so[io] + w * (2*l3 + 1) + k),
                                        (short)(wo + u * m3 + w),
                                        (short)(s1[i1] + u * (2*l1 + 1) + i),
                                        (short)(s2[i2] + j),
                                        (float)cf};
                                    T.count++;
                                }
                wo += m1 * m3;
            }
    T.wn = wo;
    // stable counting sort by output offset (O(n)) -> run-length accumulation
    // in the TP loop needs only one LDS atomic per output run.
    Table S{};
    S.count = T.count; S.wn = T.wn; S.overflow = T.overflow;
    {
        int cnt[128]{};
        for (int i = 0; i < T.count; ++i) cnt[(int)T.e[i].o]++;
        int pos[128]{};
        int acc = 0;
        for (int i = 0; i < 128; ++i) { pos[i] = acc; acc += cnt[i]; }
        for (int i = 0; i < T.count; ++i) S.e[pos[(int)T.e[i].o]++] = T.e[i];
    }
    return S;
}

constexpr Irrep IRR0_A[1] = {{16,0,1}};
constexpr Irrep IRR1_A[3] = {{16,0,1},{2,1,-1},{2,2,1}};
constexpr Irrep IRR2_A[5] = {{16,0,1},{2,1,-1},{2,1,1},{2,2,1},{2,2,-1}};
constexpr Irrep IRR3_A[8] = {{2,0,1},{2,0,-1},{2,1,1},{2,1,-1},{2,2,1},{2,2,-1},{2,3,1},{2,3,-1}};

constexpr Table HTAB1 = build_table(IRR0_A, IRR1_A);
constexpr Table HTAB2 = build_table(IRR1_A, IRR2_A);
constexpr Table HTAB3 = build_table(IRR2_A, IRR3_A);
static_assert(!HTAB1.overflow && !HTAB2.overflow && !HTAB3.overflow, "entry table overflow");

__device__ const Table g_tab1 = HTAB1;
__device__ const Table g_tab2 = HTAB2;
__device__ const Table g_tab3 = HTAB3;

constexpr int WN1 = HTAB1.wn, WN2 = HTAB2.wn, WN3 = HTAB3.wn;
constexpr int WP1 = (WN1 + 15) & ~15, WP2 = (WN2 + 15) & ~15, WP3 = (WN3 + 15) & ~15;

// ============================================================================
// WMMA helper: f32 16x16x4 (full fp32 path on the CDNA5 matrix pipe)
// A: 16x4 -> v2f per lane (lane M = l&15, K = r + 2*(l>>4))
// B: 4x16 -> v2f per lane (lane N = l&15, K = r + 2*(l>>4))
// C/D: 16x16 -> v8f per lane (M = r + 8*(l>>4), N = l&15)
// ============================================================================
__device__ inline v8f wmma_f32(v2f a, v2f b, v8f c) {
    return __builtin_amdgcn_wmma_f32_16x16x4_f32(false, a, false, b, (short)0, c,
                                                 false, false);
}

__device__ inline float softplusf(float x) {
    return fmaxf(x, 0.0f) + log1pf(expf(-fabsf(x)));
}

#ifdef HAVE_ASYNC_LDS
// Builtin wants (global int4*, LDS int4*, imm offset, imm cpol)
__device__ inline void async_b128(const float* g, float* l) {
    __builtin_amdgcn_global_load_async_to_lds_b128(
        (__attribute__((address_space(1))) v4i*)(g),
        (__attribute__((address_space(3))) v4i*)(l), 0, 0);
}
#endif

// ============================================================================
// Kernels
// ============================================================================
__global__ void zero_kernel(float* __restrict__ p, int n) {
    int i = blockIdx.x * blockDim.x + threadIdx.x;
    if (i < n) p[i] = 0.0f;
}

__global__ void sph_kernel(const float* __restrict__ vec, float* __restrict__ sh, int n) {
    int i = blockIdx.x * blockDim.x + threadIdx.x;
    if (i >= n) return;
    float x = vec[3*i + 0], y = vec[3*i + 1], z = vec[3*i + 2];
    float nn = sqrtf(x*x + y*y + z*z);
    float inv = 1.0f / fmaxf(nn, 1e-12f);
    x *= inv; y *= inv; z *= inv;
    const float s3 = 1.7320508075688772f;
    const float s5 = 2.2360679774997896f;
    const float s15 = 3.8729833462074170f;
    float* o = sh + (size_t)i * 9;
    o[0] = 1.0f;
    o[1] = s3 * y;
    o[2] = s3 * z;
    o[3] = s3 * x;
    o[4] = s15 * x * y;
    o[5] = s15 * y * z;
    o[6] = 0.5f * s5 * (3.0f * z * z - 1.0f);
    o[7] = s15 * x * z;
    o[8] = 0.5f * s15 * (x * x - y * y);
}

__global__ void count_kernel(const int* __restrict__ edge_index, float* __restrict__ cnt, int n_edges) {
    int i = blockIdx.x * blockDim.x + threadIdx.x;
    if (i < n_edges) atomicAdd(&cnt[edge_index[i]], 1.0f);
}

// h0 = node_feature(30000x256) @ lin_w(256x16) + lin_b : one wave per 16-row tile
__launch_bounds__(32)
__global__ void node_linear_kernel(const float* __restrict__ nf, const float* __restrict__ lw,
                                   const float* __restrict__ lb, float* __restrict__ h0) {
    const int lane = threadIdx.x;
    const int hi = lane >> 4;
    const int col = lane & 15;
    const int m0 = blockIdx.x * 16;
    v8f acc = {};
    for (int k0 = 0; k0 < 64; ++k0) {
        v2f a, b;
        a.x = nf[(size_t)(m0 + col) * 256 + k0*4 + 2*hi + 0];
        a.y = nf[(size_t)(m0 + col) * 256 + k0*4 + 2*hi + 1];
        b.x = lw[(size_t)(k0*4 + 2*hi + 0) * 16 + col];
        b.y = lw[(size_t)(k0*4 + 2*hi + 1) * 16 + col];
        acc = wmma_f32(a, b, acc);
    }
    float bias = lb[col];
#pragma unroll
    for (int r = 0; r < 8; ++r)
        h0[(size_t)(m0 + r + 8*hi) * 16 + col] = acc[r] + bias;
}

// Fused per-edge kernel: weight MLP (WMMA) -> tensor product (sorted CG entry
// list with run-length accumulation) -> atomic scatter-add by src.
// One wave (32 lanes) per 16-edge tile.
template <int LAYER, int IN_DIM, int OUT_DIM, int WN, int WN_PAD>
__launch_bounds__(32)
__global__ void edge_layer_kernel(const float* __restrict__ edge_feat,
                                  const float* __restrict__ edge_sh,
                                  const float* __restrict__ node_in,
                                  const int* __restrict__ edge_index,
                                  const float* __restrict__ w1, const float* __restrict__ b1,
                                  const float* __restrict__ w2, const float* __restrict__ b2,
                                  float* __restrict__ accum, int n_edges) {
    __shared__ float lds_x[16 * 64];        // edge features, then hidden after GEMM1
    __shared__ float lds_w[16 * WN_PAD];    // per-edge TP weights
    __shared__ float lds_a[16 * IN_DIM];    // gathered node[dst]
    __shared__ float lds_sh[16 * 9];        // spherical harmonics
    __shared__ float lds_out[16 * OUT_DIM]; // per-edge TP output

    const int lane = threadIdx.x;
    const int hi = lane >> 4;
    const int col = lane & 15;
    const int e0 = blockIdx.x * 16;

    const Table& T = (LAYER == 1) ? g_tab1 : (LAYER == 2) ? g_tab2 : g_tab3;

    // prefetch next tile's edge features into cache
    if (e0 + 16 < n_edges)
        __builtin_prefetch(edge_feat + (size_t)(e0 + 16) * 64 + lane * 2, 0, 1);

    // ---- stage inputs in LDS ----
#ifdef HAVE_ASYNC_LDS
    // edge features: 16 edges x 64 f32 = 256 x b128, contiguous
    for (int idx = lane; idx < 256; idx += 32)
        async_b128(edge_feat + (size_t)e0 * 64 + idx * 4, &lds_x[idx * 4]);
    // gathered node[dst] rows: IN_DIM/4 b128 chunks per edge
    {
        constexpr int CH = IN_DIM / 4;
        for (int idx = lane; idx < 16 * CH; idx += 32) {
            int e = idx / CH, ch = idx % CH;
            int dst = edge_index[n_edges + e0 + e];
            async_b128(node_in + (size_t)dst * IN_DIM + ch * 4, &lds_a[idx * 4]);
        }
    }
    // spherical harmonics: 16*9 f32 = 36 x b128, contiguous
    for (int idx = lane; idx < 36; idx += 32)
        async_b128(edge_sh + (size_t)e0 * 9 + idx * 4, &lds_sh[idx * 4]);
    for (int idx = lane; idx < 16 * OUT_DIM; idx += 32)
        lds_out[idx] = 0.0f;
    __builtin_amdgcn_s_wait_asynccnt(0);
#else
    for (int idx = lane; idx < 16 * 64; idx += 32)
        lds_x[idx] = edge_feat[(size_t)(e0 + (idx >> 6)) * 64 + (idx & 63)];
    for (int idx = lane; idx < 16 * IN_DIM; idx += 32) {
        int e = idx / IN_DIM, c = idx % IN_DIM;
        int dst = edge_index[n_edges + e0 + e];
        lds_a[idx] = node_in[(size_t)dst * IN_DIM + c];
    }
    for (int idx = lane; idx < 16 * 9; idx += 32)
        lds_sh[idx] = edge_sh[(size_t)(e0 + idx / 9) * 9 + idx % 9];
    for (int idx = lane; idx < 16 * OUT_DIM; idx += 32)
        lds_out[idx] = 0.0f;
#endif
    __syncthreads();

    // ---- GEMM1: hidden(16x64) = X(16x64) @ W1(64x64), 4 N-tiles, K-loop of 16 ----
    v8f acc[4] = {};
    for (int k0 = 0; k0 < 16; ++k0) {
        v2f af;
        af.x = lds_x[col * 64 + k0*4 + 2*hi + 0];
        af.y = lds_x[col * 64 + k0*4 + 2*hi + 1];
#pragma unroll
        for (int nt = 0; nt < 4; ++nt) {
            v2f bf;
            bf.x = w1[(size_t)(k0*4 + 2*hi + 0) * 64 + nt*16 + col];
            bf.y = w1[(size_t)(k0*4 + 2*hi + 1) * 64 + nt*16 + col];
            acc[nt] = wmma_f32(af, bf, acc[nt]);
        }
    }
    __syncthreads();
    // bias + softplus, write hidden back over lds_x
#pragma unroll
    for (int nt = 0; nt < 4; ++nt) {
        float bias = b1[nt*16 + col];
#pragma unroll
        for (int r = 0; r < 8; ++r) {
            int m = r + 8*hi;
            lds_x[m * 64 + nt*16 + col] = softplusf(acc[nt][r] + bias);
        }
    }
    __syncthreads();

    // ---- GEMM2: w(16xWN) = hidden(16x64) @ W2(64xWN) ----
    // A-fragments for the whole K range live in registers across all N tiles.
    v2f afr[16];
#pragma unroll
    for (int k0 = 0; k0 < 16; ++k0) {
        afr[k0].x = lds_x[col * 64 + k0*4 + 2*hi + 0];
        afr[k0].y = lds_x[col * 64 + k0*4 + 2*hi + 1];
    }
    constexpr int NT2 = WN_PAD / 16;
    for (int wt = 0; wt < NT2; ++wt) {
        const int n = wt * 16 + col;
        v8f a2 = {};
#pragma unroll
        for (int k0 = 0; k0 < 16; ++k0) {
            v2f bf;
            bf.x = (n < WN) ? w2[(size_t)(k0*4 + 2*hi + 0) * WN + n] : 0.0f;
            bf.y = (n < WN) ? w2[(size_t)(k0*4 + 2*hi + 1) * WN + n] : 0.0f;
            a2 = wmma_f32(afr[k0], bf, a2);
        }
        if (n < WN) {
            float bias = b2[n];
#pragma unroll
            for (int r = 0; r < 8; ++r) {
                int m = r + 8*hi;
                lds_w[m * WN_PAD + n] = a2[r] + bias;
            }
        }
    }
    __syncthreads();

    // ---- Tensor product: 2 lanes per edge, each walking a contiguous half of
    //      the out_off-sorted entry list with register run accumulation ----
    {
        const int e = col;
        const int ne = T.count;
        const int half = (ne + 1) >> 1;
        const int t0 = hi ? half : 0;
        const int t1 = hi ? ne : half;
        int cur = -1;
        float run = 0.0f;
        for (int t = t0; t < t1; ++t) {
            TPEntry en = T.e[t];
            float val = en.c * lds_w[e * WN_PAD + en.w] *
                        lds_a[e * IN_DIM + en.a] * lds_sh[e * 9 + en.s];
            if ((int)en.o != cur) {
                if (cur >= 0) atomicAdd(&lds_out[e * OUT_DIM + cur], run);
                cur = (int)en.o;
                run = 0.0f;
            }
            run += val;
        }
        if (cur >= 0) atomicAdd(&lds_out[e * OUT_DIM + cur], run);
    }
    __syncthreads();

    // ---- scatter-add into per-src-node accumulator ----
    for (int idx = lane; idx < 16 * OUT_DIM; idx += 32) {
        int e = idx / OUT_DIM, c = idx % OUT_DIM;
        int src = edge_index[e0 + e];
        atomicAdd(&accum[(size_t)src * OUT_DIM + c], lds_out[idx]);
    }
}

// mean aggregation (+ optional residual pad of previous node features)
__global__ void finalize_kernel(const float* __restrict__ accum, const float* __restrict__ cnt,
                                const float* __restrict__ resid, int resid_dim,
                                float* __restrict__ out, int total, int dim) {
    int idx = blockIdx.x * blockDim.x + threadIdx.x;
    if (idx >= total) return;
    int n = idx / dim, c = idx % dim;
    float v = accum[idx] / fmaxf(cnt[n], 1.0f);
    if (resid != nullptr && c < resid_dim) v += resid[(size_t)n * resid_dim + c];
    out[idx] = v;
}

// ============================================================================
// Workspace layout (floats). Zeroed span = [OFF_CNT, WS_TOTAL).
// ============================================================================
constexpr size_t OFF_SH  = 0;
constexpr size_t OFF_H0  = OFF_SH + (size_t)N_EDGES * 9;
constexpr size_t OFF_L1  = OFF_H0 + (size_t)N_NODES * 16;
constexpr size_t OFF_L2  = OFF_L1 + (size_t)N_NODES * 32;
constexpr size_t OFF_CNT = OFF_L2 + (size_t)N_NODES * 48;
constexpr size_t OFF_A1  = OFF_CNT + (size_t)N_NODES;
constexpr size_t OFF_A2  = OFF_A1 + (size_t)N_NODES * 32;
constexpr size_t OFF_A3  = OFF_A2 + (size_t)N_NODES * 48;
constexpr size_t WS_TOTAL = OFF_A3 + (size_t)N_NODES * 64;
constexpr size_t ZERO_SPAN = WS_TOTAL - OFF_CNT;

extern "C" void kernel_launch(void* const* d_in, const int* in_sizes, int n_in,
                              void* d_out, int out_size, void* d_ws, size_t ws_size,
                              hipStream_t stream) {
    const float* nf   = (const float*)d_in[0];
    const float* ev   = (const float*)d_in[1];
    const float* ef   = (const float*)d_in[2];
    const int*   ei   = (const int*)d_in[3];
    const float* lw   = (const float*)d_in[4];
    const float* lb   = (const float*)d_in[5];
    const float* f1w1 = (const float*)d_in[6];
    const float* f1b1 = (const float*)d_in[7];
    const float* f1w2 = (const float*)d_in[8];
    const float* f1b2 = (const float*)d_in[9];
    const float* f2w1 = (const float*)d_in[10];
    const float* f2b1 = (const float*)d_in[11];
    const float* f2w2 = (const float*)d_in[12];
    const float* f2b2 = (const float*)d_in[13];
    const float* f3w1 = (const float*)d_in[14];
    const float* f3b1 = (const float*)d_in[15];
    const float* f3w2 = (const float*)d_in[16];
    const float* f3b2 = (const float*)d_in[17];

    float* ws   = (float*)d_ws;
    float* sh   = ws + OFF_SH;
    float* h0   = ws + OFF_H0;
    float* l1   = ws + OFF_L1;
    float* l2   = ws + OFF_L2;
    float* cnt  = ws + OFF_CNT;
    float* a1   = ws + OFF_A1;
    float* a2   = ws + OFF_A2;
    float* a3   = ws + OFF_A3;
    float* outp = (float*)d_out;

    {
        int n = (int)ZERO_SPAN;
        zero_kernel<<<(n + 255) / 256, 256, 0, stream>>>(ws + OFF_CNT, n);
    }
    sph_kernel<<<(N_EDGES + 255) / 256, 256, 0, stream>>>(ev, sh, N_EDGES);
    node_linear_kernel<<<N_NODES / 16, 32, 0, stream>>>(nf, lw, lb, h0);
    count_kernel<<<(N_EDGES + 255) / 256, 256, 0, stream>>>(ei, cnt, N_EDGES);

    const int eblocks = N_EDGES / 16;

    edge_layer_kernel<1, 16, 32, WN1, WP1><<<eblocks, 32, 0, stream>>>(
        ef, sh, h0, ei, f1w1, f1b1, f1w2, f1b2, a1, N_EDGES);
    finalize_kernel<<<(N_NODES * 32 + 255) / 256, 256, 0, stream>>>(
        a1, cnt, h0, 16, l1, N_NODES * 32, 32);

    edge_layer_kernel<2, 32, 48, WN2, WP2><<<eblocks, 32, 0, stream>>>(
        ef, sh, l1, ei, f2w1, f2b1, f2w2, f2b2, a2, N_EDGES);
    finalize_kernel<<<(N_NODES * 48 + 255) / 256, 256, 0, stream>>>(
        a2, cnt, nullptr, 0, l2, N_NODES * 48, 48);

    edge_layer_kernel<3, 48, 64, WN3, WP3><<<eblocks, 32, 0, stream>>>(
        ef, sh, l2, ei, f3w1, f3b1, f3w2, f3b2, a3, N_EDGES);
    finalize_kernel<<<(N_NODES * 64 + 255) / 256, 256, 0, stream>>>(
        a3, cnt, nullptr, 0, outp, N_NODES * 64, 64);

    (void)in_sizes; (void)n_in; (void)out_size; (void)ws_size;
}